// ModelNetShapeLoss_25074019074111
// MI455X (gfx1250) — compile-verified
//
#include <hip/hip_runtime.h>
#include <math.h>

// ---------------- Problem constants (from setup_inputs) ----------------
#define BATCH  16
#define NPTS   2048          // N == M
#define NTILES (NPTS / 16)   // 128
#define LFLAT  (NPTS * 3)    // 6144 values per batch for EMD sort
#define SORT_N 8192          // next pow2 >= 6144

typedef __attribute__((ext_vector_type(2))) float v2f;
typedef __attribute__((ext_vector_type(8))) float v8f;

// acc layout in workspace:
// acc[0] = sum over (b,n) of min_m d(pred,gt)
// acc[1] = sum over (b,m) of min_n d(pred,gt)   (rows of gt vs pred)
// acc[2] = sum over all sorted elems of (xs-ys)^2
// acc[3] = sum over (b,n) of exp(-5*min self-dist)
// acc[4] = sum over b of unbiased var of centroid distances

__global__ void init_acc_kernel(float* acc) {
    if (threadIdx.x < 8) acc[threadIdx.x] = 0.0f;
}

// Distance tile via V_WMMA_F32_16X16X4_F32 with the K=4 fold:
// A' = (-2x0,-2x1,-2x2,1), B'_m = (y0,y1,y2,|y_m|^2)  =>  tile = |y|^2 - 2 x.y
__device__ __forceinline__ v8f dist_tile(const float* __restrict__ Yb, int mb,
                                         int lane, float sel, v2f a) {
    const int col = mb + (lane & 15);
    const float* yc = Yb + (size_t)col * 3;
    const float y0 = yc[0], y1 = yc[1], y2c = yc[2];
    const float y2n = y0 * y0 + y1 * y1 + y2c * y2c;
    v2f bmat;                                   // branch-free K-striped build
    bmat.x = y2c + sel * (y0 - y2c);            // K=0 | K=2
    bmat.y = y2n + sel * (y1 - y2n);            // K=1 | K=3(|y|^2)
    v8f c = {0.f, 0.f, 0.f, 0.f, 0.f, 0.f, 0.f, 0.f};
    return __builtin_amdgcn_wmma_f32_16x16x4_f32(
               false, a, false, bmat, (short)0, c, false, false);
}

// One wave per 16-row tile of X; loop over all 16-col tiles of Y.
// blockIdx.y selects mode: 0 = pred-vs-gt (acc0), 1 = gt-vs-pred (acc1),
// 2 = self with +1e6 diagonal, accumulate exp(-5*min) (acc3).
// The diagonal tile (SELF only) is skipped in the main loop via a scalar
// index remap and peeled afterwards, so the hot loop is branch-free.
__global__ void pc_min_wmma_kernel(const float* __restrict__ pred,
                                   const float* __restrict__ gt,
                                   float* __restrict__ acc) {
    const int mode = blockIdx.y;                    // scalar
    const float* X = (mode == 1) ? gt : pred;
    const float* Y = (mode == 0) ? gt : pred;
    const int accIdx = (mode == 0) ? 0 : (mode == 1) ? 1 : 3;
    const bool SELF = (mode == 2);

    const int lane = threadIdx.x & 31;
    // Wave-scalar tile coordinates (s_* ops, scalar addresses).
    const int wib    = __builtin_amdgcn_readfirstlane(threadIdx.x >> 5);
    const int waveId = blockIdx.x * 8 + wib;
    const int b  = waveId / NTILES;                 // scalar
    const int rb = (waveId % NTILES) << 4;          // scalar row base

    // ---- A operand (loop-invariant): rows rb..rb+15, K striped ----
    const int row = rb + (lane & 15);
    const float* xr = X + ((size_t)b * NPTS + row) * 3;
    const float x0 = xr[0], x1 = xr[1], x2c = xr[2];
    const float myx2 = x0 * x0 + x1 * x1 + x2c * x2c;
    const float sel = (lane < 16) ? 1.0f : 0.0f;    // branch-free selector
    v2f a;
    a.x = -2.0f * (x2c + sel * (x0 - x2c));         // K=0 | K=2
    a.y = (lane < 16) ? (-2.0f * x1) : 1.0f;        // K=1 | K=3(=1)

    // Broadcast |x_row|^2 for the 8 C-rows each half-wave owns.
    float x2sel[8];
#pragma unroll
    for (int r = 0; r < 8; ++r) {
        const float lo = __shfl(myx2, r, 32);
        const float hi = __shfl(myx2, r + 8, 32);
        x2sel[r] = (lane < 16) ? lo : hi;
    }

    float rm[8];
#pragma unroll
    for (int r = 0; r < 8; ++r) rm[r] = 1.0e30f;

    const float* Yb = Y + (size_t)b * NPTS * 3;

    // Scalar remap skips the diagonal tile in SELF mode (skipIdx==NTILES
    // otherwise, so nothing is skipped and all 128 tiles are visited).
    const int skipIdx = SELF ? (rb >> 4) : NTILES;
    const int iters   = SELF ? (NTILES - 1) : NTILES;
#pragma unroll 2
    for (int t = 0; t < iters; ++t) {
        const int tt = t + ((t >= skipIdx) ? 1 : 0);   // s_cselect
        const v8f c = dist_tile(Yb, tt << 4, lane, sel, a);
#pragma unroll
        for (int r = 0; r < 8; ++r)
            rm[r] = fminf(rm[r], c[r] + x2sel[r]);     // single path -> min3
    }

    if (SELF) {
        // Peeled diagonal tile (once, cold): +1e6 on the diagonal (EYE_BIG).
        // Diagonal hits VGPR r when (lane&15) - 8*(lane>=16) == r.
        const int diagR = (lane & 15) - ((lane >> 4) << 3);
        const v8f c = dist_tile(Yb, rb, lane, sel, a);
#pragma unroll
        for (int r = 0; r < 8; ++r) {
            float d = c[r] + x2sel[r];
            if (diagR == r) d += 1.0e6f;
            rm[r] = fminf(rm[r], d);
        }
    }

    // Row-min across the 16 lanes sharing a row (both halves in parallel),
    // then sum f(min) over the 16 rows of this tile.
    float s = 0.0f;
#pragma unroll
    for (int r = 0; r < 8; ++r) {
        float v = rm[r];
        v = fminf(v, __shfl_xor(v, 1, 32));
        v = fminf(v, __shfl_xor(v, 2, 32));
        v = fminf(v, __shfl_xor(v, 4, 32));
        v = fminf(v, __shfl_xor(v, 8, 32));
        s += SELF ? __expf(-5.0f * v) : v;
    }
    s += __shfl_xor(s, 16, 32);                     // combine the two row halves
    if (lane == 0) atomicAdd(&acc[accIdx], s);
}

// Bitonic sort of one batch's flattened cloud (6144 vals padded to 8192) in LDS.
__global__ void bitonic_sort_kernel(const float* __restrict__ pred,
                                    const float* __restrict__ gt,
                                    float* __restrict__ sortedP,
                                    float* __restrict__ sortedG) {
    __shared__ float s[SORT_N];
    const int b     = blockIdx.x >> 1;
    const int which = blockIdx.x & 1;
    const float* src = which ? gt : pred;
    float* dst       = which ? sortedG : sortedP;
    const size_t base = (size_t)b * LFLAT;

    for (int i = threadIdx.x; i < SORT_N; i += blockDim.x)
        s[i] = (i < LFLAT) ? src[base + i] : 3.4e38f;
    __syncthreads();

    for (int k = 2; k <= SORT_N; k <<= 1) {
        for (int j = k >> 1; j > 0; j >>= 1) {
            for (int i = threadIdx.x; i < SORT_N; i += blockDim.x) {
                const int ixj = i ^ j;
                if (ixj > i) {
                    const float va = s[i], vb = s[ixj];
                    const bool up = ((i & k) == 0);
                    if ((va > vb) == up) { s[i] = vb; s[ixj] = va; }
                }
            }
            __syncthreads();
        }
    }
    for (int i = threadIdx.x; i < LFLAT; i += blockDim.x)
        dst[base + i] = s[i];
}

__global__ void emd_diff_kernel(const float* __restrict__ xs,
                                const float* __restrict__ ys,
                                float* __restrict__ acc, int accIdx) {
    __shared__ float red[256];
    const int total = BATCH * LFLAT;
    float p = 0.0f;
    for (int i = blockIdx.x * blockDim.x + threadIdx.x; i < total;
         i += gridDim.x * blockDim.x) {
        const float d = xs[i] - ys[i];
        p += d * d;
    }
    red[threadIdx.x] = p;
    __syncthreads();
    for (int o = 128; o > 0; o >>= 1) {
        if ((int)threadIdx.x < o) red[threadIdx.x] += red[threadIdx.x + o];
        __syncthreads();
    }
    if (threadIdx.x == 0) atomicAdd(&acc[accIdx], red[0]);
}

// Unbiased variance of distances-to-centroid, one block per batch.
__global__ void centroid_var_kernel(const float* __restrict__ pc,
                                    float* __restrict__ acc, int accIdx) {
    __shared__ float r0[256], r1[256], r2[256];
    __shared__ float cx, cy, cz;
    const int b = blockIdx.x;
    const float* p = pc + (size_t)b * NPTS * 3;

    float sx = 0.f, sy = 0.f, sz = 0.f;
    for (int n = threadIdx.x; n < NPTS; n += blockDim.x) {
        sx += p[n * 3 + 0]; sy += p[n * 3 + 1]; sz += p[n * 3 + 2];
    }
    r0[threadIdx.x] = sx; r1[threadIdx.x] = sy; r2[threadIdx.x] = sz;
    __syncthreads();
    for (int o = 128; o > 0; o >>= 1) {
        if ((int)threadIdx.x < o) {
            r0[threadIdx.x] += r0[threadIdx.x + o];
            r1[threadIdx.x] += r1[threadIdx.x + o];
            r2[threadIdx.x] += r2[threadIdx.x + o];
        }
        __syncthreads();
    }
    if (threadIdx.x == 0) {
        cx = r0[0] / NPTS; cy = r1[0] / NPTS; cz = r2[0] / NPTS;
    }
    __syncthreads();

    float sd = 0.f, sd2 = 0.f;
    for (int n = threadIdx.x; n < NPTS; n += blockDim.x) {
        const float dx = p[n * 3 + 0] - cx;
        const float dy = p[n * 3 + 1] - cy;
        const float dz = p[n * 3 + 2] - cz;
        const float dist = sqrtf(dx * dx + dy * dy + dz * dz);
        sd += dist; sd2 += dist * dist;
    }
    r0[threadIdx.x] = sd; r1[threadIdx.x] = sd2;
    __syncthreads();
    for (int o = 128; o > 0; o >>= 1) {
        if ((int)threadIdx.x < o) {
            r0[threadIdx.x] += r0[threadIdx.x + o];
            r1[threadIdx.x] += r1[threadIdx.x + o];
        }
        __syncthreads();
    }
    if (threadIdx.x == 0) {
        const float var = (r1[0] - r0[0] * r0[0] / (float)NPTS) / (float)(NPTS - 1);
        atomicAdd(&acc[accIdx], var);
    }
}

__global__ void finalize_kernel(const float* __restrict__ acc,
                                float* __restrict__ out) {
    const float chamfer = acc[0] / (float)(BATCH * NPTS)
                        + acc[1] / (float)(BATCH * NPTS);
    const float emd = acc[2] / (float)(BATCH * LFLAT);
    const float nn  = acc[3] / (float)(BATCH * NPTS);
    const float var = acc[4] / (float)BATCH;
    out[0] = 1.0f * chamfer + 0.5f * emd + 0.3f * (var + nn);
}

extern "C" void kernel_launch(void* const* d_in, const int* in_sizes, int n_in,
                              void* d_out, int out_size, void* d_ws, size_t ws_size,
                              hipStream_t stream) {
    const float* pred = (const float*)d_in[0];   // (16,2048,3) f32
    const float* gt   = (const float*)d_in[1];   // (16,2048,3) f32
    float* out = (float*)d_out;

    // workspace layout
    float* acc     = (float*)d_ws;                                   // 8 floats
    float* sortedP = (float*)((char*)d_ws + 256);                    // B*L floats
    float* sortedG = sortedP + (size_t)BATCH * LFLAT;                // B*L floats

    init_acc_kernel<<<1, 32, 0, stream>>>(acc);

    // Fused: chamfer row-mins (both directions) + self-NN penalty via blockIdx.y.
    const int waves  = BATCH * NTILES;           // 2048 waves per mode
    const int blocks = waves / 8;                // 8 waves / 256-thread block
    pc_min_wmma_kernel<<<dim3(blocks, 3), 256, 0, stream>>>(pred, gt, acc);

    // EMD: per-batch bitonic sorts then squared diff
    bitonic_sort_kernel<<<2 * BATCH, 256, 0, stream>>>(pred, gt, sortedP, sortedG);
    emd_diff_kernel<<<128, 256, 0, stream>>>(sortedP, sortedG, acc, 2);

    // consistency variance term
    centroid_var_kernel<<<BATCH, 256, 0, stream>>>(pred, acc, 4);

    finalize_kernel<<<1, 1, 0, stream>>>(acc, out);
}